// Activation1d_74947179316159
// MI455X (gfx1250) — compile-verified
//
#include <hip/hip_runtime.h>
#include <stdint.h>

// ---------------------------------------------------------------------------
// Anti-aliased snake activation (up2 -> snake -> down2), K=12, fp32.
//   B=16, C=512, T=8192.  Memory-bound (~512MB traffic -> ~22us @ 23.3TB/s).
//   Stage 1: async global->LDS x tile.   Stage 2: sliding-window up+snake.
//   Stage 3: downsample as banded matmul, 11x V_WMMA_F32_16X16X4_F32 per
//            16x16 output tile (fp32 accumulate, matches reference math).
// ---------------------------------------------------------------------------

#define T_LEN      8192
#define C_LEN      512
#define ROWS_BLK   16      // channels (rows) per workgroup
#define TOUT_TILE  256     // output time samples per workgroup
#define XCOLS      268     // x halo tile: [t0-6, t0+261]
#define XPITCH     272     // padded (slide-ahead reads up to col 268)
#define ACT_COLS   524     // act tile: u in [2*t0-6, 2*t0+517] (clamp folded in)
#define ACT_PITCH  524     // 524 % 64 banks -> conflict-free 16-row gathers
#define NPAIR      262     // ACT_COLS/2 act-sample pairs per row
#define PAIRS_PER  17      // ceil(262/16) contiguous pairs per thread

typedef float v2f __attribute__((ext_vector_type(2)));
typedef float v8f __attribute__((ext_vector_type(8)));

typedef __attribute__((address_space(1))) int as1_int;
typedef __attribute__((address_space(3))) int as3_int;

#if __has_builtin(__builtin_amdgcn_global_load_async_to_lds_b32) && \
    __has_builtin(__builtin_amdgcn_s_wait_asynccnt)
#define USE_ASYNC_LDS 1
#warning "CDNA5 probe: async global->LDS builtins AVAILABLE (using them)"
#else
#warning "CDNA5 probe: async global->LDS builtins NOT available (fallback path)"
#endif

__global__ __launch_bounds__(256) void antialias_snake_kernel(
    const float* __restrict__ x,
    const float* __restrict__ up_f,    // 12 taps
    const float* __restrict__ dn_f,    // 12 taps
    const float* __restrict__ alpha,   // [C]
    const float* __restrict__ beta,    // [C]
    float* __restrict__ out)
{
    __shared__ __align__(16) float xs[ROWS_BLK * XPITCH];
    __shared__ __align__(16) float acts[ROWS_BLK * ACT_PITCH];

    const int tid  = threadIdx.x;
    const int row  = tid >> 4;            // 0..15
    const int l16  = tid & 15;
    const int rb   = blockIdx.y;          // row block (B*C/16 = 512)
    const int t0   = blockIdx.x * TOUT_TILE;
    const int row_g = rb * ROWS_BLK + row;          // flattened (b,c) row
    const long xrow = (long)row_g * T_LEN;
    const int xb   = t0 - 6;                         // x tile global base col

    // ---- stage 1: x tile (with edge clamp) -> LDS ------------------------
    {
        const float* src = x + xrow;
        for (int col = l16; col < XCOLS; col += 16) {
            int g = xb + col;
            g = g < 0 ? 0 : (g > T_LEN - 1 ? T_LEN - 1 : g);
#if defined(USE_ASYNC_LDS)
            __builtin_amdgcn_global_load_async_to_lds_b32(
                (as1_int*)(uintptr_t)(src + g),
                (as3_int*)(uint32_t)(uintptr_t)&xs[row * XPITCH + col],
                0, 0);
#else
            xs[row * XPITCH + col] = src[g];
#endif
        }
    }
#if defined(USE_ASYNC_LDS)
    __builtin_amdgcn_s_wait_asynccnt(0);
#endif
    __syncthreads();

    // ---- stage 2: upsample + snake -> act tile (sliding-window pairs) ----
    // up[2i+p] = sum_{j=0..5} (2*f_up[2j+1-p]) * x[clamp(i+2+p-j)]
    // act(u)   = up + (1/exp(beta)) * sin(up*exp(alpha))^2, u index-clamped.
    // Thread owns PAIRS_PER contiguous (even,odd) pairs; x window in regs:
    // w[k] = x[clamp(i-3+k)] = xs_row[pq + k], since (i - xb) == pq + 3.
    {
        const int ch = row_g & (C_LEN - 1);
        const float a_coef = __expf(alpha[ch]);
        const float binv   = 1.0f / (__expf(beta[ch]) + 1e-9f);

        float fu[12];
#pragma unroll
        for (int k = 0; k < 12; ++k) fu[k] = 2.0f * up_f[k];  // fold UP_RATIO

        const float* xl = &xs[row * XPITCH];
        float* al       = &acts[row * ACT_PITCH];
        const int ubase = 2 * t0 - 6;

        // generic (edge) evaluation with the u-clamp folded in
        auto up_at = [&](int u) -> float {
            u = u < 0 ? 0 : (u > 2 * T_LEN - 1 ? 2 * T_LEN - 1 : u);
            const int i = u >> 1, p = u & 1;
            const int base = i + 2 + p - xb;
            if (p)
                return fu[0]  * xl[base]     + fu[2]  * xl[base - 1]
                     + fu[4]  * xl[base - 2] + fu[6]  * xl[base - 3]
                     + fu[8]  * xl[base - 4] + fu[10] * xl[base - 5];
            return fu[1]  * xl[base]     + fu[3]  * xl[base - 1]
                 + fu[5]  * xl[base - 2] + fu[7]  * xl[base - 3]
                 + fu[9]  * xl[base - 4] + fu[11] * xl[base - 5];
        };

        const int pq0 = PAIRS_PER * l16;
        const int pqe = (pq0 + PAIRS_PER) < NPAIR ? (pq0 + PAIRS_PER) : NPAIR;

        float w0 = xl[pq0],     w1 = xl[pq0 + 1], w2 = xl[pq0 + 2],
              w3 = xl[pq0 + 3], w4 = xl[pq0 + 4], w5 = xl[pq0 + 5],
              w6 = xl[pq0 + 6];

#pragma unroll 4
        for (int pq = pq0; pq < pqe; ++pq) {
            const int u0 = ubase + 2 * pq;      // even phase sample index
            float up0, up1;
            if (u0 >= 0 && u0 <= 2 * T_LEN - 2) {   // fast interior path
                up0 = fu[1] * w5 + fu[3] * w4 + fu[5]  * w3
                    + fu[7] * w2 + fu[9] * w1 + fu[11] * w0;
                up1 = fu[0] * w6 + fu[2] * w5 + fu[4]  * w4
                    + fu[6] * w3 + fu[8] * w2 + fu[10] * w1;
            } else {                                 // edge tiles only
                up0 = up_at(u0);
                up1 = up_at(u0 + 1);
            }
            const float s0 = __sinf(up0 * a_coef);
            const float s1 = __sinf(up1 * a_coef);
            float2 o;
            o.x = fmaf(s0 * s0, binv, up0);
            o.y = fmaf(s1 * s1, binv, up1);
            *reinterpret_cast<float2*>(&al[2 * pq]) = o;   // 8B-aligned
            w0 = w1; w1 = w2; w2 = w3; w3 = w4; w4 = w5; w5 = w6;
            w6 = xl[pq + 7];   // next window top (pad reads stay < XPITCH)
        }
    }
    __syncthreads();

    // ---- stage 3: downsample as banded matmul via WMMA f32 16x16x4 -------
    // out[m][n] = sum_uu A[m][uu] * W[uu][n],  A = act window (16 x 44),
    // W[uu][n] = f_dn[uu - 2n - 1] (banded), 11 K=4 chunks accumulated in C.
    const int wid  = tid >> 5;     // wave id 0..7
    const int lane = tid & 31;
    const int n    = lane & 15;    // B/D column (output time within tile)
    const int half = lane >> 4;    // lane group

    // B fragments, branch-free (clamped-index load + select). Constant
    // across tiles & waves. Assumed 4x16 f32 B layout (mirror of A):
    // lanes 0-15 VGPR0 = K row 2*half, VGPR1 = K row 2*half+1, col = lane&15.
    v2f bfr[11];
#pragma unroll
    for (int c0 = 0; c0 < 11; ++c0) {
        const int i0 = 4 * c0 + 2 * half - 2 * n - 1;
        const int i1 = i0 + 1;
        const int i0c = i0 < 0 ? 0 : (i0 > 11 ? 11 : i0);
        const int i1c = i1 < 0 ? 0 : (i1 > 11 ? 11 : i1);
        const float v0 = dn_f[i0c];
        const float v1 = dn_f[i1c];
        v2f b;
        b.x = (i0 == i0c) ? v0 : 0.0f;
        b.y = (i1 == i1c) ? v1 : 0.0f;
        bfr[c0] = b;
    }

    const int m = lane & 15;       // A row = channel within block
#pragma unroll
    for (int tt = 0; tt < 2; ++tt) {
        const int jt = wid + 8 * tt;          // 16-wide output tile 0..15
        const int wb = 2 * (jt * 16);         // act window base col
        v8f acc = {};
#pragma unroll
        for (int c0 = 0; c0 < 11; ++c0) {
            // A fragment: lane holds A[m][2*half + {0,1}] of this K=4 chunk
            const float2 av = *reinterpret_cast<const float2*>(
                &acts[m * ACT_PITCH + wb + 4 * c0 + 2 * half]);
            v2f a; a.x = av.x; a.y = av.y;
            acc = __builtin_amdgcn_wmma_f32_16x16x4_f32(
                false, a, false, bfr[c0], (short)0, acc, false, false);
        }
        // D layout: VGPR v -> row v (lanes 0-15) / v+8 (lanes 16-31), col n.
        const int tcol = t0 + jt * 16 + n;
#pragma unroll
        for (int v = 0; v < 8; ++v) {
            const int mrow = v + 8 * half;
            out[(long)(rb * ROWS_BLK + mrow) * T_LEN + tcol] = acc[v];
        }
    }
}

extern "C" void kernel_launch(void* const* d_in, const int* in_sizes, int n_in,
                              void* d_out, int out_size, void* d_ws, size_t ws_size,
                              hipStream_t stream) {
    const float* x   = (const float*)d_in[0];
    const float* upf = (const float*)d_in[1];
    const float* dnf = (const float*)d_in[2];
    const float* al  = (const float*)d_in[3];
    const float* be  = (const float*)d_in[4];
    float* out = (float*)d_out;

    dim3 grid(T_LEN / TOUT_TILE, (16 * C_LEN) / ROWS_BLK);  // (32, 512)
    antialias_snake_kernel<<<grid, 256, 0, stream>>>(x, upf, dnf, al, be, out);
}